// GraphNN_214748364910
// MI455X (gfx1250) — compile-verified
//
#include <hip/hip_runtime.h>

typedef __attribute__((ext_vector_type(2))) float v2f;
typedef __attribute__((ext_vector_type(8))) float v8f;

#define BN_EPS 1e-5f

// ---------------------------------------------------------------- fill
__global__ void fill_kernel(float* __restrict__ p, float v, long n) {
    long i = (long)blockIdx.x * blockDim.x + threadIdx.x;
    if (i < n) p[i] = v;
}

// ---------------------------------------------------------------- degree
__global__ void deg_kernel(float* __restrict__ deg, const int* __restrict__ dst, int E) {
    int e = blockIdx.x * blockDim.x + threadIdx.x;
    if (e < E) atomicAdd(&deg[dst[e]], 1.0f);
}

__global__ void rsqrt_kernel(float* __restrict__ d, int n) {
    int i = blockIdx.x * blockDim.x + threadIdx.x;
    if (i < n) d[i] = rsqrtf(d[i]);   // deg >= 1 always (self-loop)
}

// ---------------------------------------------------------------- GEMM via V_WMMA_F32_16X16X4_F32
// C[M,Nc] = A[M,K] @ B[K,Nc].  One wave32 computes one 16x16 tile of C.
// A frag 16x4 f32: lanes 0-15 hold rows M=0..15, kh=lane>>4 selects K half:
//   elem j -> A[m, k + 2*kh + j].  B frag 4x16 is the mirror (per-lane column).
// C/D: VGPR v at lane l -> C[v + 8*(l>>4), l&15].
__global__ void gemm_wmma_kernel(const float* __restrict__ A, const float* __restrict__ B,
                                 float* __restrict__ C, int M, int K, int Nc) {
    int lane = threadIdx.x & 31;
    int wid  = (blockIdx.x * blockDim.x + threadIdx.x) >> 5;
    int tiles_n = Nc >> 4;
    int tiles_m = M >> 4;
    if (wid >= tiles_m * tiles_n) return;
    int tm = wid / tiles_n, tn = wid % tiles_n;
    int mr = lane & 15;      // A row / B,C column within tile
    int kh = lane >> 4;      // K-half selector

    const float* Ap = A + (long)(tm * 16 + mr) * K + 2 * kh;
    const float* Bp = B + (long)(2 * kh) * Nc + tn * 16 + mr;

    v8f c = {};
    for (int k = 0; k < K; k += 4) {
        v2f a = *(const v2f*)(Ap + k);
        v2f b;
        b.x = Bp[(long)k * Nc];
        b.y = Bp[(long)k * Nc + Nc];
        c = __builtin_amdgcn_wmma_f32_16x16x4_f32(false, a, false, b,
                                                  (short)0, c, false, false);
    }

    float* Cp = C + (long)(tm * 16 + 8 * kh) * Nc + tn * 16 + mr;
#pragma unroll
    for (int v = 0; v < 8; ++v)
        Cp[(long)v * Nc] = c[v];
}

// ---------------------------------------------------------------- edge scatter
// thread t: edge e = t >> cpe_log2, chunk c = t & (cpe-1); 4 floats per chunk.
// Lanes of a wave share an edge (F=128) or split it in halves (F=64):
// h0[src] gathers are fully coalesced float4 loads; scatter via f32 atomics (L2-resident).
__global__ void scatter_kernel(const float* __restrict__ h, const int* __restrict__ src,
                               const int* __restrict__ dst, const float* __restrict__ dinv,
                               float* __restrict__ agg, int E, int F, int cpe_log2) {
    long t = (long)blockIdx.x * blockDim.x + threadIdx.x;
    long e = t >> cpe_log2;
    int  c = (int)(t & (((long)1 << cpe_log2) - 1));
    if (e >= E) return;
    int s = src[e], d = dst[e];
    float w = dinv[s] * dinv[d];
    const float4 v = *(const float4*)(h + (long)s * F + 4 * c);
    float* out = agg + (long)d * F + 4 * c;
    atomicAdd(out + 0, w * v.x);
    atomicAdd(out + 1, w * v.y);
    atomicAdd(out + 2, w * v.z);
    atomicAdd(out + 3, w * v.w);
}

// ---------------------------------------------------------------- self-loop + bias
__global__ void selfloop_bias_kernel(const float* __restrict__ h, const float* __restrict__ dinv,
                                     const float* __restrict__ bias, float* __restrict__ agg,
                                     long total, int Fmask, int Flog2) {
    long t = (long)blockIdx.x * blockDim.x + threadIdx.x;
    if (t >= total) return;
    int  f = (int)(t & Fmask);
    long i = t >> Flog2;
    float di = dinv[i];
    agg[t] += di * di * h[t] + bias[f];
}

// ---------------------------------------------------------------- BN stats (sum, sumsq per feature)
__global__ void bn_stats_kernel(const float* __restrict__ x, float* __restrict__ sums,
                                float* __restrict__ sumsq, int n, int F) {
    __shared__ float s1[256];
    __shared__ float s2[256];
    int tid = threadIdx.x;
    int f = tid & (F - 1);
    int g = tid / F;
    int groups = blockDim.x / F;
    float a = 0.f, b = 0.f;
    for (long i = (long)blockIdx.x * groups + g; i < n; i += (long)gridDim.x * groups) {
        float v = x[i * F + f];
        a += v; b += v * v;
    }
    s1[tid] = a; s2[tid] = b;
    __syncthreads();
    if (tid < F) {
        for (int gg = 1; gg < groups; ++gg) { a += s1[tid + gg * F]; b += s2[tid + gg * F]; }
        atomicAdd(&sums[f], a);
        atomicAdd(&sumsq[f], b);
    }
}

// ---------------------------------------------------------------- BN apply (in place)
__global__ void bn_apply_kernel(float* __restrict__ x, const float* __restrict__ sums,
                                const float* __restrict__ sumsq, const float* __restrict__ gamma,
                                const float* __restrict__ beta, long total, int Fmask, float invN) {
    long t = (long)blockIdx.x * blockDim.x + threadIdx.x;
    if (t >= total) return;
    int f = (int)(t & Fmask);
    float mean = sums[f] * invN;
    float var  = sumsq[f] * invN - mean * mean;
    x[t] = (x[t] - mean) * rsqrtf(var + BN_EPS) * gamma[f] + beta[f];
}

// ---------------------------------------------------------------- launch
extern "C" void kernel_launch(void* const* d_in, const int* in_sizes, int n_in,
                              void* d_out, int out_size, void* d_ws, size_t ws_size,
                              hipStream_t stream) {
    (void)n_in; (void)out_size; (void)ws_size;
    const float* emb   = (const float*)d_in[0];
    const float* W1    = (const float*)d_in[1];
    const float* b1    = (const float*)d_in[2];
    const float* W2    = (const float*)d_in[3];
    const float* b2    = (const float*)d_in[4];
    const float* gamma = (const float*)d_in[5];
    const float* beta  = (const float*)d_in[6];
    const int*   ei    = (const int*)d_in[7];

    const int F2 = in_sizes[4];        // ninp = 64
    const int F1 = in_sizes[2];        // 2*ninp = 128
    const int N  = in_sizes[0] / F2;   // 50000 (= 3125 * 16)
    const int E  = in_sizes[7] / 2;    // 800000
    const int* src = ei;
    const int* dst = ei + E;

    float* ws    = (float*)d_ws;
    float* dinv  = ws;                          // [N]   (deg -> dinv in place)
    float* hbuf  = ws + N;                      // [N*F1] h0, later reused for h1
    float* agg1  = hbuf + (long)N * F1;         // [N*F1]
    float* sums  = agg1 + (long)N * F1;         // [F2]
    float* sumsq = sums + F2;                   // [F2]
    float* out   = (float*)d_out;               // doubles as agg2

    const dim3 blk(256);
    auto cdiv = [](long a, long b) { return (unsigned)((a + b - 1) / b); };

    // init: deg=1 (self-loop), accumulators = 0
    fill_kernel<<<cdiv(N, 256), blk, 0, stream>>>(dinv, 1.0f, N);
    fill_kernel<<<cdiv((long)N * F1, 256), blk, 0, stream>>>(agg1, 0.0f, (long)N * F1);
    fill_kernel<<<cdiv((long)N * F2, 256), blk, 0, stream>>>(out, 0.0f, (long)N * F2);
    fill_kernel<<<1, blk, 0, stream>>>(sums, 0.0f, 2 * F2);

    // degrees -> dinv
    deg_kernel<<<cdiv(E, 256), blk, 0, stream>>>(dinv, dst, E);
    rsqrt_kernel<<<cdiv(N, 256), blk, 0, stream>>>(dinv, N);

    // layer 1: h0 = X @ W1  (M=N, K=F2, Nc=F1)
    {
        long waves = (long)(N / 16) * (F1 / 16);
        gemm_wmma_kernel<<<cdiv(waves * 32, 256), blk, 0, stream>>>(emb, W1, hbuf, N, F2, F1);
    }
    scatter_kernel<<<cdiv((long)E * (F1 >> 2), 256), blk, 0, stream>>>(
        hbuf, src, dst, dinv, agg1, E, F1, 5 /* 128/4 = 32 chunks */);
    selfloop_bias_kernel<<<cdiv((long)N * F1, 256), blk, 0, stream>>>(
        hbuf, dinv, b1, agg1, (long)N * F1, F1 - 1, 7);

    // layer 2: h1 = agg1 @ W2  (M=N, K=F1, Nc=F2); h1 reuses hbuf
    {
        long waves = (long)(N / 16) * (F2 / 16);
        gemm_wmma_kernel<<<cdiv(waves * 32, 256), blk, 0, stream>>>(agg1, W2, hbuf, N, F1, F2);
    }
    scatter_kernel<<<cdiv((long)E * (F2 >> 2), 256), blk, 0, stream>>>(
        hbuf, src, dst, dinv, out, E, F2, 4 /* 64/4 = 16 chunks */);
    selfloop_bias_kernel<<<cdiv((long)N * F2, 256), blk, 0, stream>>>(
        hbuf, dinv, b2, out, (long)N * F2, F2 - 1, 6);

    // batchnorm over nodes
    bn_stats_kernel<<<512, blk, 0, stream>>>(out, sums, sumsq, N, F2);
    bn_apply_kernel<<<cdiv((long)N * F2, 256), blk, 0, stream>>>(
        out, sums, sumsq, gamma, beta, (long)N * F2, F2 - 1, 1.0f / (float)N);
}